// MoE_64312840290787
// MI455X (gfx1250) — compile-verified
//
#include <hip/hip_runtime.h>
#include <math.h>
#include <stdint.h>

// ---------------- Static problem config (matches reference) ----------------
#define NTOK 8192      // N = B*T
#define DDIM 1024      // D
#define NEXP 8         // E
#define HDIM 1408      // H
#define CAP  2560      // CAPACITY
#define AUXC 0.01f

// LDS tile strides padded so row-stride % 64 == 4 (16 rows hit 16 distinct
// banks) while keeping 16B alignment (stride*4 % 16 == 0) for async b128.
#define XS_STRIDE 1028
#define HS_STRIDE 1412
#define SMEM_BYTES ((16 * XS_STRIDE + 16 * HS_STRIDE) * 4)  // 156,160 B

typedef float v2f __attribute__((ext_vector_type(2)));
typedef float v8f __attribute__((ext_vector_type(8)));

__device__ __forceinline__ v8f wmma_f32(v2f a, v2f b, v8f c) {
  // 8-arg pattern: (neg_a, A, neg_b, B, c_mod, C, reuse_a, reuse_b)
  return __builtin_amdgcn_wmma_f32_16x16x4_f32(false, a, false, b, (short)0, c,
                                               false, false);
}

// Async global->LDS 16B copy (per-lane addresses), tracked by ASYNCcnt.
__device__ __forceinline__ void async_copy_b128(const float* src, float* lds_dst) {
  unsigned dstoff = (unsigned)(uintptr_t)lds_dst;  // generic LDS addr[31:0]
  asm volatile("global_load_async_to_lds_b128 %0, %1, off"
               :: "v"(dstoff), "v"(src) : "memory");
}
__device__ __forceinline__ void wait_async0() {
  asm volatile("s_wait_asynccnt 0x0" ::: "memory");
}

// ---------------- Workspace init ----------------
__global__ void moe_init_kernel(int* __restrict__ ecount, int* __restrict__ top1,
                                float* __restrict__ psum,
                                int* __restrict__ slot_token,
                                float* __restrict__ slot_prob) {
  int i = blockIdx.x * blockDim.x + threadIdx.x;
  if (i < NEXP) { ecount[i] = 0; top1[i] = 0; psum[i] = 0.f; }
  for (int s = i; s < NEXP * CAP; s += gridDim.x * blockDim.x) {
    slot_token[s] = 0;
    slot_prob[s] = 0.f;
  }
}

// ---------------- Router: logits, softmax, top-2, dispatch, aux accum -------
// One wave (32 lanes) per token. blockDim = 256 -> 8 tokens / block.
__global__ void moe_router_kernel(const float* __restrict__ x,
                                  const float* __restrict__ rw,
                                  int* __restrict__ ecount,
                                  int* __restrict__ top1cnt,
                                  float* __restrict__ psum,
                                  int* __restrict__ slot_token,
                                  float* __restrict__ slot_prob) {
  const int wave = threadIdx.x >> 5;
  const int lane = threadIdx.x & 31;
  const int n = blockIdx.x * (blockDim.x >> 5) + wave;
  if (n >= NTOK) return;

  float acc[NEXP];
#pragma unroll
  for (int e = 0; e < NEXP; e++) acc[e] = 0.f;

  const float* xr = x + (size_t)n * DDIM;
  for (int d = lane; d < DDIM; d += 32) {
    float xv = xr[d];
#pragma unroll
    for (int e = 0; e < NEXP; e++) acc[e] += xv * rw[e * DDIM + d];
  }
#pragma unroll
  for (int e = 0; e < NEXP; e++) {
#pragma unroll
    for (int off = 16; off >= 1; off >>= 1)
      acc[e] += __shfl_xor(acc[e], off, 32);
  }

  if (lane == 0) {
    float mx = acc[0];
#pragma unroll
    for (int e = 1; e < NEXP; e++) mx = fmaxf(mx, acc[e]);
    float p[NEXP], s = 0.f;
#pragma unroll
    for (int e = 0; e < NEXP; e++) { p[e] = __expf(acc[e] - mx); s += p[e]; }
    float inv = 1.f / s;
#pragma unroll
    for (int e = 0; e < NEXP; e++) { p[e] *= inv; atomicAdd(&psum[e], p[e]); }

    int i1 = 0;
#pragma unroll
    for (int e = 1; e < NEXP; e++) if (p[e] > p[i1]) i1 = e;
    int i2 = (i1 == 0) ? 1 : 0;
#pragma unroll
    for (int e = 0; e < NEXP; e++) if (e != i1 && p[e] > p[i2]) i2 = e;

    atomicAdd(&top1cnt[i1], 1);

    float renorm = 1.f / (p[i1] + p[i2] + 1e-9f);
    int eidx[2] = {i1, i2};
    float pk[2] = {p[i1] * renorm, p[i2] * renorm};
#pragma unroll
    for (int k = 0; k < 2; k++) {
      int e = eidx[k];
      int pos = atomicAdd(&ecount[e], 1);
      if (pos < CAP) {
        slot_token[e * CAP + pos] = n;
        slot_prob[e * CAP + pos] = pk[k];
      }
    }
  }
}

// ---------------- Aux loss scalar ----------------
__global__ void moe_aux_kernel(const int* __restrict__ top1cnt,
                               const float* __restrict__ psum,
                               float* __restrict__ out_aux) {
  if (threadIdx.x == 0 && blockIdx.x == 0) {
    float a = 0.f;
#pragma unroll
    for (int e = 0; e < NEXP; e++)
      a += ((float)top1cnt[e] / (float)NTOK) * (psum[e] / (float)NTOK);
    *out_aux = AUXC * (float)NEXP * a;
  }
}

// ---------------- Fused expert MLP: relu(X*Wfc)^2 * Wproj ----------------
// One block = 16 rows of one expert. 256 threads = 8 waves.
// X tile staged to LDS with GLOBAL_LOAD_ASYNC_TO_LDS_B128 (ASYNCcnt path).
// Stage 1: each wave produces 176 H-cols of the 16x1408 H tile (LDS).
// Stage 2: each wave produces 128 D-cols of the 16x1024 output tile.
// ROUTED=0: identity rows (shared expert), plain store to y.
// ROUTED=1: rows gathered via slot_token, atomicAdd(prob * out) into y.
template <int ROUTED>
__global__ __launch_bounds__(256) void moe_mlp_kernel(
    const float* __restrict__ x,
    const float* __restrict__ wfc_all,
    const float* __restrict__ wproj_all,
    const int* __restrict__ slot_token,
    const float* __restrict__ slot_prob,
    const int* __restrict__ ecount,
    float* __restrict__ y) {
  extern __shared__ float smem[];
  float* Xs = smem;                    // 16 x XS_STRIDE
  float* Hs = smem + 16 * XS_STRIDE;   // 16 x HS_STRIDE

  const int e = blockIdx.y;
  const int row0 = blockIdx.x * 16;
  int count;
  if (ROUTED) {
    count = ecount[e];
    if (count > CAP) count = CAP;
  } else {
    count = NTOK;
  }
  if (row0 >= count) return;  // uniform across block

  const float* wfc = wfc_all + (size_t)e * DDIM * HDIM;
  const float* wproj = wproj_all + (size_t)e * HDIM * DDIM;
  const int tid = threadIdx.x;

  // ---- Gather X row tile directly into LDS via async b128 copies ----
  // 16 rows x 256 float4 = 4096 lane-transfers; 16 per thread.
  for (int idx = tid; idx < 16 * (DDIM / 4); idx += blockDim.x) {
    int r = idx >> 8;        // DDIM/4 == 256
    int c4 = idx & 255;
    int tok = ROUTED ? slot_token[e * CAP + row0 + r] : (row0 + r);
    const float* src = x + (size_t)tok * DDIM + c4 * 4;
    async_copy_b128(src, Xs + r * XS_STRIDE + c4 * 4);
  }
  wait_async0();     // this wave's async transfers landed in LDS
  __syncthreads();   // all waves' transfers visible

  const int wv = tid >> 5;
  const int lane = tid & 31;
  const int lm = lane & 15;   // A: row M / B,C,D: col N
  const int hi = lane >> 4;   // half-wave selector (K or M offset)

  // ---- Stage 1: H = relu(X @ Wfc)^2 into LDS ----
  {
    const int hbase = wv * 176;  // 8 waves * 176 = 1408
    v8f c[11] = {};
    for (int k = 0; k < DDIM; k += 4) {
      v2f a;
      a.x = Xs[lm * XS_STRIDE + k + 2 * hi];
      a.y = Xs[lm * XS_STRIDE + k + 2 * hi + 1];
      const float* bp = wfc + (size_t)(k + 2 * hi) * HDIM + hbase + lm;
#pragma unroll
      for (int t = 0; t < 11; t++) {
        v2f b;
        b.x = bp[t * 16];
        b.y = bp[t * 16 + HDIM];
        c[t] = wmma_f32(a, b, c[t]);
      }
    }
#pragma unroll
    for (int t = 0; t < 11; t++) {
#pragma unroll
      for (int r = 0; r < 8; r++) {
        float v = c[t][r];
        v = v > 0.f ? v * v : 0.f;  // relu(x)^2
        Hs[(r + 8 * hi) * HS_STRIDE + hbase + t * 16 + lm] = v;
      }
    }
  }
  __syncthreads();

  // ---- Stage 2: Out = H @ Wproj ----
  {
    const int nbase = wv * 128;  // 8 waves * 128 = 1024
    v8f d[8] = {};
    for (int k = 0; k < HDIM; k += 4) {
      v2f a;
      a.x = Hs[lm * HS_STRIDE + k + 2 * hi];
      a.y = Hs[lm * HS_STRIDE + k + 2 * hi + 1];
      const float* bp = wproj + (size_t)(k + 2 * hi) * DDIM + nbase + lm;
#pragma unroll
      for (int t = 0; t < 8; t++) {
        v2f b;
        b.x = bp[t * 16];
        b.y = bp[t * 16 + DDIM];
        d[t] = wmma_f32(a, b, d[t]);
      }
    }
#pragma unroll
    for (int r = 0; r < 8; r++) {
      int row = r + 8 * hi;
      int grow = row0 + row;
      if (grow < count) {
        if (ROUTED) {
          int tok = slot_token[e * CAP + grow];
          float p = slot_prob[e * CAP + grow];
          float* yr = y + (size_t)tok * DDIM + nbase + lm;
#pragma unroll
          for (int t = 0; t < 8; t++) atomicAdd(&yr[t * 16], p * d[t][r]);
        } else {
          float* yr = y + (size_t)grow * DDIM + nbase + lm;
#pragma unroll
          for (int t = 0; t < 8; t++) yr[t * 16] = d[t][r];
        }
      }
    }
  }
}

// ---------------- Host launcher ----------------
extern "C" void kernel_launch(void* const* d_in, const int* in_sizes, int n_in,
                              void* d_out, int out_size, void* d_ws,
                              size_t ws_size, hipStream_t stream) {
  (void)in_sizes; (void)n_in; (void)out_size; (void)ws_size;
  const float* x       = (const float*)d_in[0];  // [N, D]
  const float* rw      = (const float*)d_in[1];  // [E, D]
  const float* w_fc    = (const float*)d_in[2];  // [E, D, H]
  const float* w_proj  = (const float*)d_in[3];  // [E, H, D]
  const float* ws_fc   = (const float*)d_in[4];  // [1, D, H]
  const float* ws_proj = (const float*)d_in[5];  // [1, H, D]

  float* y = (float*)d_out;                            // [N, D] flat
  float* aux = ((float*)d_out) + (size_t)NTOK * DDIM;  // trailing scalar

  // Workspace layout (all 4-byte aligned)
  int* ecount      = (int*)d_ws;                        // [E]
  int* top1        = ecount + NEXP;                     // [E]
  float* psum      = (float*)(top1 + NEXP);             // [E]
  int* slot_token  = (int*)(psum + NEXP);               // [E*CAP]
  float* slot_prob = (float*)(slot_token + NEXP * CAP); // [E*CAP]

  // 1) init workspace
  moe_init_kernel<<<dim3(81), dim3(256), 0, stream>>>(ecount, top1, psum,
                                                      slot_token, slot_prob);
  // 2) router + dispatch + aux accumulation (1024 blocks * 8 tokens)
  moe_router_kernel<<<dim3(NTOK / 8), dim3(256), 0, stream>>>(
      x, rw, ecount, top1, psum, slot_token, slot_prob);
  // 3) aux loss scalar
  moe_aux_kernel<<<dim3(1), dim3(32), 0, stream>>>(top1, psum, aux);
  // 4) shared expert: stores y (covers every element -> no pre-zero needed)
  moe_mlp_kernel<0><<<dim3(NTOK / 16, 1), dim3(256), SMEM_BYTES, stream>>>(
      x, ws_fc, ws_proj, nullptr, nullptr, nullptr, y);
  // 5) routed experts: atomic accumulate prob-weighted outputs into y
  moe_mlp_kernel<1><<<dim3(CAP / 16, NEXP), dim3(256), SMEM_BYTES, stream>>>(
      x, w_fc, w_proj, slot_token, slot_prob, ecount, y);
}